// GAT_5995774345731
// MI455X (gfx1250) — compile-verified
//
#include <hip/hip_runtime.h>
#include <hip/hip_bf16.h>

typedef __attribute__((ext_vector_type(2))) float v2f;
typedef __attribute__((ext_vector_type(8))) float v8f;

#define NEG_SLOPE 0.2f

// ---------- float ordered-uint encoding for atomicMax on floats ----------
__device__ __forceinline__ unsigned f2ord(float f) {
  unsigned b = __float_as_uint(f);
  return (b & 0x80000000u) ? ~b : (b | 0x80000000u);
}
__device__ __forceinline__ float ord2f(unsigned k) {
  unsigned b = (k & 0x80000000u) ? (k & 0x7FFFFFFFu) : ~k;
  return __uint_as_float(b);
}

__global__ void zero_f32(float* __restrict__ p, long long n) {
  long long i = (long long)blockIdx.x * blockDim.x + threadIdx.x;
  if (i < n) p[i] = 0.0f;
}

// ================= GEMM 1: h1[N,128] = x[N,128] @ W[128,128] =============
// One wave -> one 16x16 C tile. Block = 256 threads = 8 waves = 8 col tiles.
// WMMA f32 16x16x4: A lanes 0-15 hold K={k0,k0+1}, lanes 16-31 K={k0+2,k0+3};
// C VGPR r: M = r + 8*(lane/16), N = lane%16.
__global__ void gemm1_wmma(const float* __restrict__ x, const float* __restrict__ W,
                           float* __restrict__ out) {
  const int tile = blockIdx.x;           // 16 nodes per tile (N divisible by 16)
  const int wave = threadIdx.x >> 5;     // 0..7 -> 16-wide column tile
  const int lane = threadIdx.x & 31;
  const int half = lane >> 4;
  const int l16  = lane & 15;
  const long long row = (long long)tile * 16 + l16;
  const int col = wave * 16 + l16;
  v8f c = {};
#pragma unroll
  for (int k0 = 0; k0 < 128; k0 += 4) {
    const int ka = k0 + 2 * half;
    v2f a, b;
    a[0] = x[row * 128 + ka];
    a[1] = x[row * 128 + ka + 1];
    b[0] = W[(long long)ka * 128 + col];
    b[1] = W[(long long)(ka + 1) * 128 + col];
    c = __builtin_amdgcn_wmma_f32_16x16x4_f32(false, a, false, b, (short)0, c,
                                              false, false);
  }
#pragma unroll
  for (int r = 0; r < 8; ++r) {
    const int m = r + 8 * half;
    out[((long long)tile * 16 + m) * 128 + col] = c[r];
  }
}

// ============ GEMM 2: z[N,40] = hmid[N,128] @ W[128,40] (pad to 48) =======
// Block = 96 threads = 3 waves = 3 column tiles (cols 40..47 are zero-padded).
__global__ void gemm2_wmma(const float* __restrict__ hmid, const float* __restrict__ W,
                           float* __restrict__ z) {
  const int tile = blockIdx.x;
  const int wave = threadIdx.x >> 5;     // 0..2
  const int lane = threadIdx.x & 31;
  const int half = lane >> 4;
  const int l16  = lane & 15;
  const long long row = (long long)tile * 16 + l16;
  const int col = wave * 16 + l16;                 // 0..47 logical
  const int cc  = (col < 40) ? col : 39;           // in-bounds address
  const float msk = (col < 40) ? 1.0f : 0.0f;      // value select, no EXEC change
  v8f c = {};
#pragma unroll
  for (int k0 = 0; k0 < 128; k0 += 4) {
    const int ka = k0 + 2 * half;
    v2f a, b;
    a[0] = hmid[row * 128 + ka];
    a[1] = hmid[row * 128 + ka + 1];
    b[0] = W[ka * 40 + cc] * msk;
    b[1] = W[(ka + 1) * 40 + cc] * msk;
    c = __builtin_amdgcn_wmma_f32_16x16x4_f32(false, a, false, b, (short)0, c,
                                              false, false);
  }
  if (col < 40) {
#pragma unroll
    for (int r = 0; r < 8; ++r) {
      const int m = r + 8 * half;
      z[((long long)tile * 16 + m) * 40 + col] = c[r];
    }
  }
}

// ============ attention scalar projections ============
// layer1: i = n*8 + h; h1 row chunk base = (n*8+h)*16
__global__ void attn1(const float* __restrict__ h1, const float* __restrict__ att_src,
                      const float* __restrict__ att_dst, float* __restrict__ asrc,
                      float* __restrict__ adst, long long NH) {
  long long i = (long long)blockIdx.x * blockDim.x + threadIdx.x;
  if (i >= NH) return;
  const int hd = (int)(i & 7);
  const float* hp = h1 + i * 16;
  const float* as = att_src + hd * 16;
  const float* ad = att_dst + hd * 16;
  float ds = 0.f, dd = 0.f;
#pragma unroll
  for (int c = 0; c < 16; ++c) {
    float v = hp[c];
    ds += v * as[c];
    dd += v * ad[c];
  }
  asrc[i] = ds;
  adst[i] = dd;
}

__global__ void attn2(const float* __restrict__ z, const float* __restrict__ att_src,
                      const float* __restrict__ att_dst, float* __restrict__ asrc,
                      float* __restrict__ adst, long long N) {
  long long n = (long long)blockIdx.x * blockDim.x + threadIdx.x;
  if (n >= N) return;
  float ds = 0.f, dd = 0.f;
  const float* zp = z + n * 40;
#pragma unroll
  for (int c = 0; c < 40; ++c) {
    float v = zp[c];
    ds += v * att_src[c];
    dd += v * att_dst[c];
  }
  asrc[n] = ds;
  adst[n] = dd;
}

// ============ edge passes, layer 1 (8 heads) ============
__device__ __forceinline__ void edge_sd(const long long* ei, long long E, long long e,
                                        long long& s, long long& d) {
  if (e < E) { s = ei[e]; d = ei[E + e]; } else { s = e - E; d = s; }  // self loops
}

__global__ void edge_max1(const long long* __restrict__ ei, long long E, long long N,
                          const float* __restrict__ asrc, const float* __restrict__ adst,
                          unsigned* __restrict__ m) {
  long long e = (long long)blockIdx.x * blockDim.x + threadIdx.x;
  if (e >= E + N) return;
  long long s, d; edge_sd(ei, E, e, s, d);
#pragma unroll
  for (int h = 0; h < 8; ++h) {
    float v = asrc[s * 8 + h] + adst[d * 8 + h];
    v = (v > 0.f) ? v : NEG_SLOPE * v;
    atomicMax(&m[d * 8 + h], f2ord(v));
  }
}

__global__ void edge_den1(const long long* __restrict__ ei, long long E, long long N,
                          const float* __restrict__ asrc, const float* __restrict__ adst,
                          const unsigned* __restrict__ m, float* __restrict__ den) {
  long long e = (long long)blockIdx.x * blockDim.x + threadIdx.x;
  if (e >= E + N) return;
  long long s, d; edge_sd(ei, E, e, s, d);
#pragma unroll
  for (int h = 0; h < 8; ++h) {
    float v = asrc[s * 8 + h] + adst[d * 8 + h];
    v = (v > 0.f) ? v : NEG_SLOPE * v;
    atomicAdd(&den[d * 8 + h], __expf(v - ord2f(m[d * 8 + h])));
  }
}

// one thread per (edge, head): gather 16 src features, scatter-add alpha*feat
__global__ void edge_agg1(const long long* __restrict__ ei, long long E, long long N,
                          const float* __restrict__ asrc, const float* __restrict__ adst,
                          const unsigned* __restrict__ m, const float* __restrict__ den,
                          const float* __restrict__ h1, float* __restrict__ agg) {
  long long i = (long long)blockIdx.x * blockDim.x + threadIdx.x;
  if (i >= (E + N) * 8) return;
  long long e = i >> 3;
  const int h = (int)(i & 7);
  long long s, d; edge_sd(ei, E, e, s, d);
  float v = asrc[s * 8 + h] + adst[d * 8 + h];
  v = (v > 0.f) ? v : NEG_SLOPE * v;
  const float alpha = __expf(v - ord2f(m[d * 8 + h])) / (den[d * 8 + h] + 1e-16f);
  const float4* hs = (const float4*)(h1 + s * 128 + h * 16);
  float* o = agg + d * 128 + h * 16;
#pragma unroll
  for (int q = 0; q < 4; ++q) {
    float4 p = hs[q];
    atomicAdd(o + 4 * q + 0, alpha * p.x);
    atomicAdd(o + 4 * q + 1, alpha * p.y);
    atomicAdd(o + 4 * q + 2, alpha * p.z);
    atomicAdd(o + 4 * q + 3, alpha * p.w);
  }
}

// bias + ELU, in place (agg1 -> hmid)
__global__ void bias_elu(float* __restrict__ a, const float* __restrict__ b,
                         long long total) {
  long long i = (long long)blockIdx.x * blockDim.x + threadIdx.x;
  if (i >= total) return;
  const int c = (int)(i & 127);
  float v = a[i] + b[c];
  a[i] = (v > 0.f) ? v : expm1f(v);
}

// ============ edge passes, layer 2 (1 head, 40 ch) ============
__global__ void edge_max2(const long long* __restrict__ ei, long long E, long long N,
                          const float* __restrict__ asrc, const float* __restrict__ adst,
                          unsigned* __restrict__ m) {
  long long e = (long long)blockIdx.x * blockDim.x + threadIdx.x;
  if (e >= E + N) return;
  long long s, d; edge_sd(ei, E, e, s, d);
  float v = asrc[s] + adst[d];
  v = (v > 0.f) ? v : NEG_SLOPE * v;
  atomicMax(&m[d], f2ord(v));
}

__global__ void edge_den2(const long long* __restrict__ ei, long long E, long long N,
                          const float* __restrict__ asrc, const float* __restrict__ adst,
                          const unsigned* __restrict__ m, float* __restrict__ den) {
  long long e = (long long)blockIdx.x * blockDim.x + threadIdx.x;
  if (e >= E + N) return;
  long long s, d; edge_sd(ei, E, e, s, d);
  float v = asrc[s] + adst[d];
  v = (v > 0.f) ? v : NEG_SLOPE * v;
  atomicAdd(&den[d], __expf(v - ord2f(m[d])));
}

__global__ void edge_agg2(const long long* __restrict__ ei, long long E, long long N,
                          const float* __restrict__ asrc, const float* __restrict__ adst,
                          const unsigned* __restrict__ m, const float* __restrict__ den,
                          const float* __restrict__ z, float* __restrict__ agg) {
  long long e = (long long)blockIdx.x * blockDim.x + threadIdx.x;
  if (e >= E + N) return;
  long long s, d; edge_sd(ei, E, e, s, d);
  float v = asrc[s] + adst[d];
  v = (v > 0.f) ? v : NEG_SLOPE * v;
  const float alpha = __expf(v - ord2f(m[d])) / (den[d] + 1e-16f);
  const float* zs = z + s * 40;
  float* o = agg + d * 40;
#pragma unroll
  for (int c = 0; c < 40; ++c) atomicAdd(o + c, alpha * zs[c]);
}

// bias + log_softmax over 40 classes
__global__ void final_lsm(const float* __restrict__ agg, const float* __restrict__ b,
                          float* __restrict__ out, long long N) {
  long long n = (long long)blockIdx.x * blockDim.x + threadIdx.x;
  if (n >= N) return;
  float v[40];
  float mx = -3.402823466e38f;
#pragma unroll
  for (int c = 0; c < 40; ++c) {
    v[c] = agg[n * 40 + c] + b[c];
    mx = fmaxf(mx, v[c]);
  }
  float ssum = 0.f;
#pragma unroll
  for (int c = 0; c < 40; ++c) ssum += __expf(v[c] - mx);
  const float l = __logf(ssum);
#pragma unroll
  for (int c = 0; c < 40; ++c) out[n * 40 + c] = v[c] - mx - l;
}

static inline unsigned blks(long long n, int b) { return (unsigned)((n + b - 1) / b); }

extern "C" void kernel_launch(void* const* d_in, const int* in_sizes, int n_in,
                              void* d_out, int out_size, void* d_ws, size_t ws_size,
                              hipStream_t stream) {
  const float*     x     = (const float*)d_in[0];
  const long long* ei    = (const long long*)d_in[1];   // int64 [2,E]
  const float*     W1    = (const float*)d_in[2];
  const float*     as1   = (const float*)d_in[3];
  const float*     ad1   = (const float*)d_in[4];
  const float*     b1    = (const float*)d_in[5];
  const float*     W2    = (const float*)d_in[6];
  const float*     as2   = (const float*)d_in[7];
  const float*     ad2   = (const float*)d_in[8];
  const float*     b2    = (const float*)d_in[9];
  float*           out   = (float*)d_out;

  const long long N = in_sizes[0] / 128;   // 100000 (divisible by 16)
  const long long E = in_sizes[1] / 2;     // 1600000
  const long long T = E + N;               // edges incl. self loops

  // ---- workspace layout (floats) ----
  float* ws    = (float*)d_ws;
  float* h1    = ws;                         // N*128
  float* agg1  = h1 + N * 128;               // N*128 (becomes hmid in place)
  float* asrc1 = agg1 + N * 128;             // N*8
  float* adst1 = asrc1 + N * 8;              // N*8
  unsigned* m1 = (unsigned*)(adst1 + N * 8); // N*8
  float* den1  = (float*)(m1 + N * 8);       // N*8
  // layer-2 reuse (h1 region free after edge_agg1; small arrays reused):
  float* z2    = h1;                         // N*40
  float* agg2  = h1 + N * 40;                // N*40 (disjoint from z2)
  float* asrc2 = asrc1;                      // N
  float* adst2 = adst1;                      // N
  unsigned* m2 = m1;                         // N
  float* den2  = den1;                       // N

  // ---- layer 1 ----
  zero_f32<<<blks(N * 128, 256), 256, 0, stream>>>(agg1, N * 128);
  zero_f32<<<blks(N * 16, 256), 256, 0, stream>>>((float*)m1, N * 16); // m1+den1 contiguous

  gemm1_wmma<<<(unsigned)(N / 16), 256, 0, stream>>>(x, W1, h1);
  attn1<<<blks(N * 8, 256), 256, 0, stream>>>(h1, as1, ad1, asrc1, adst1, N * 8);

  edge_max1<<<blks(T, 256), 256, 0, stream>>>(ei, E, N, asrc1, adst1, m1);
  edge_den1<<<blks(T, 256), 256, 0, stream>>>(ei, E, N, asrc1, adst1, m1, den1);
  edge_agg1<<<blks(T * 8, 256), 256, 0, stream>>>(ei, E, N, asrc1, adst1, m1, den1,
                                                  h1, agg1);
  bias_elu<<<blks(N * 128, 256), 256, 0, stream>>>(agg1, b1, N * 128);

  // ---- layer 2 ----
  zero_f32<<<blks(N * 40, 256), 256, 0, stream>>>(agg2, N * 40);
  zero_f32<<<blks(N, 256), 256, 0, stream>>>((float*)m2, N);
  zero_f32<<<blks(N, 256), 256, 0, stream>>>(den2, N);

  gemm2_wmma<<<(unsigned)(N / 16), 96, 0, stream>>>(agg1 /*hmid*/, W2, z2);
  attn2<<<blks(N, 256), 256, 0, stream>>>(z2, as2, ad2, asrc2, adst2, N);

  edge_max2<<<blks(T, 256), 256, 0, stream>>>(ei, E, N, asrc2, adst2, m2);
  edge_den2<<<blks(T, 256), 256, 0, stream>>>(ei, E, N, asrc2, adst2, m2, den2);
  edge_agg2<<<blks(T, 256), 256, 0, stream>>>(ei, E, N, asrc2, adst2, m2, den2,
                                              z2, agg2);

  final_lsm<<<blks(N, 256), 256, 0, stream>>>(agg2, b2, out, N);
}